// RoPEDynamicCacheAttention_39779987096359
// MI455X (gfx1250) — compile-verified
//
#include <hip/hip_runtime.h>
#include <hip/hip_bf16.h>

// ---- shapes ----
#define Bq   8
#define Tq   4
#define HID  4096
#define Hh   32
#define Dd   128
#define SC   4096
#define SF   4100   // SC + Tq
#define NROW 32     // B*T

typedef __attribute__((ext_vector_type(2))) float v2f;
typedef __attribute__((ext_vector_type(8))) float v8f;

static __device__ inline v8f wmma_f32(v2f a, v2f b, v8f c) {
  // V_WMMA_F32_16X16X4_F32 : D = A(16x4) * B(4x16) + C(16x16)
  return __builtin_amdgcn_wmma_f32_16x16x4_f32(
      /*neg_a=*/false, a, /*neg_b=*/false, b,
      /*c_mod=*/(short)0, c, /*reuse_a=*/false, /*reuse_b=*/false);
}

static __device__ inline float rowmax16(float x) {
  for (int m = 1; m < 16; m <<= 1) x = fmaxf(x, __shfl_xor(x, m, 32));
  return x;
}
static __device__ inline float rowsum16(float x) {
  for (int m = 1; m < 16; m <<= 1) x += __shfl_xor(x, m, 32);
  return x;
}

// ---------------------------------------------------------------------------
// GEMM: dst[m][n] = sum_k A[m][k] * W[n][k]   (A: [32][4096], W: [4096][4096])
// one wave per 16-wide N-tile, computes both 16-row M-tiles so each W element
// is streamed exactly once (non-temporal).  dst_mode 0: row-major [32][4096];
// dst_mode 1: scatter into v_full tail.
// ---------------------------------------------------------------------------
__global__ __launch_bounds__(256) void gemm_nt(const float* __restrict__ A,
                                               const float* __restrict__ W,
                                               float* __restrict__ dst,
                                               int dst_mode) {
  const int tid  = threadIdx.x;
  const int wave = tid >> 5;
  const int lane = tid & 31;
  const int lg   = lane >> 4;      // 0: K=0,1   1: K=2,3
  const int ln   = lane & 15;
  const int ntile = blockIdx.x * 8 + wave;     // 0..255
  const int koff  = 2 * lg;

  const float* a0p = A + (long)ln * HID;        // rows 0..15
  const float* a1p = A + (long)(16 + ln) * HID; // rows 16..31
  const float* wp  = W + (long)(ntile * 16 + ln) * HID;

  v8f c0 = {0.f,0.f,0.f,0.f,0.f,0.f,0.f,0.f};
  v8f c1 = {0.f,0.f,0.f,0.f,0.f,0.f,0.f,0.f};

  for (int k0 = 0; k0 < HID; k0 += 4) {
    if ((k0 & 255) == 0) __builtin_prefetch(wp + k0 + 1024, 0, 0);
    v2f a0 = *(const v2f*)(a0p + k0 + koff);
    v2f a1 = *(const v2f*)(a1p + k0 + koff);
    v2f b  = __builtin_nontemporal_load((const v2f*)(wp + k0 + koff));
    c0 = wmma_f32(a0, b, c0);
    c1 = wmma_f32(a1, b, c1);
  }

  const int n = ntile * 16 + ln;
  for (int r = 0; r < 8; ++r) {
    int M0 = r + 8 * lg;     // 0..15
    int M1 = 16 + M0;        // 16..31
    if (dst_mode == 0) {
      dst[(long)M0 * HID + n] = c0[r];
      dst[(long)M1 * HID + n] = c1[r];
    } else {
      // scatter into v_full tail: [b][h][4096+t][d]
      int h = n >> 7, d = n & 127;
      {
        int b = M0 >> 2, t = M0 & 3;
        dst[((long)(b * Hh + h) * SF + SC + t) * Dd + d] = c0[r];
      }
      {
        int b = M1 >> 2, t = M1 & 3;
        dst[((long)(b * Hh + h) * SF + SC + t) * Dd + d] = c1[r];
      }
    }
  }
}

// ---------------------------------------------------------------------------
// RoPE on q (in place) and k (written to k_full tail).
// ---------------------------------------------------------------------------
__global__ __launch_bounds__(256) void rope_append(float* __restrict__ q,
                                                   const float* __restrict__ kraw,
                                                   const float* __restrict__ cosb,
                                                   const float* __restrict__ sinb,
                                                   float* __restrict__ kfull) {
  int idx = blockIdx.x * blockDim.x + threadIdx.x;   // 0..65535
  int m   = idx >> 11;
  int rem = idx & 2047;
  int h   = rem >> 6;
  int d   = rem & 63;
  int col0 = h * Dd + d, col1 = col0 + 64;

  float c0 = cosb[m * Dd + d],      s0 = sinb[m * Dd + d];
  float c1 = cosb[m * Dd + d + 64], s1 = sinb[m * Dd + d + 64];

  float q0 = q[(long)m * HID + col0], q1 = q[(long)m * HID + col1];
  q[(long)m * HID + col0] = q0 * c0 - q1 * s0;
  q[(long)m * HID + col1] = q1 * c1 + q0 * s1;

  float k0 = kraw[(long)m * HID + col0], k1 = kraw[(long)m * HID + col1];
  int b = m >> 2, t = m & 3;
  long base = ((long)(b * Hh + h) * SF + SC + t) * Dd;
  kfull[base + d]      = k0 * c0 - k1 * s0;
  kfull[base + d + 64] = k1 * c1 + k0 * s1;
}

// ---------------------------------------------------------------------------
// Fused flash attention + KV-cache materialization.
// One block per (b,h); 8 waves split S.  Every cache element is read exactly
// once from the inputs (NT), fed to WMMA, and stored once (NT) into the
// k_full/v_full outputs -- the standalone copy pass (and its 1.07 GB of
// re-reads) is eliminated.  Tail rows (s>=4096, RoPE'd K / projected V) were
// already written to the outputs and are only read here.
// ---------------------------------------------------------------------------
#define NW 8
__global__ __launch_bounds__(256) void attn_fused(const float* __restrict__ qrope,
                                                  const float* __restrict__ cache_k,
                                                  const float* __restrict__ cache_v,
                                                  float* __restrict__ kfull,
                                                  float* __restrict__ vfull,
                                                  float* __restrict__ attn_out) {
  __shared__ float qA[16 * 132];        // padded: conflict-free column reads
  __shared__ float pS[NW][16 * 17];     // per-wave P staging (C->A relayout)
  __shared__ float mS[NW][4];
  __shared__ float lS[NW][4];
  __shared__ float accS[NW][4 * 128];   // only the 4 real query rows

  const int bh = blockIdx.x;
  const int b = bh >> 5, h = bh & 31;
  const int tid = threadIdx.x;
  const int wave = tid >> 5, lane = tid & 31;
  const int lg = lane >> 4, ln = lane & 15;
  const int koff = 2 * lg;
  const float scale = 0.08838834764831845f;   // 1/sqrt(128)

  // stage scaled Q (rows 4..15 zero-padded)
  for (int idx = tid; idx < 16 * 128; idx += 256) {
    int r = idx >> 7, d = idx & 127;
    float v = 0.f;
    if (r < Tq) v = qrope[((long)(b * Tq + r)) * HID + h * Dd + d] * scale;
    qA[r * 132 + d] = v;
  }
  __syncthreads();

  const float* Kc = cache_k + (long)bh * SC * Dd;   // cache input  [4096][128]
  const float* Vc = cache_v + (long)bh * SC * Dd;
  float* Kf = kfull + (long)bh * SF * Dd;           // output       [4100][128]
  float* Vf = vfull + (long)bh * SF * Dd;

  float mr[4], lr[4];
  v8f o[8];
  for (int r = 0; r < 4; ++r) { mr[r] = -3.0e38f; lr[r] = 0.f; }
  for (int j = 0; j < 8; ++j) o[j] = (v8f){0.f,0.f,0.f,0.f,0.f,0.f,0.f,0.f};

  const int ntiles = (SF + 15) / 16;   // 257
  for (int ci = wave; ci < ntiles; ci += NW) {
    const int s0 = ci * 16;

    // ---- scores tile: Q(16x128) . K^T, streaming K through registers ----
    int srow = s0 + ln;
    const bool ktail = srow >= SC;
    if (srow > SF - 1) srow = SF - 1;
    const float* kr = (ktail ? (const float*)Kf : Kc) + (long)srow * Dd;
    float* kw = Kf + (long)srow * Dd;

    v8f c = (v8f){0.f,0.f,0.f,0.f,0.f,0.f,0.f,0.f};
    for (int k0 = 0; k0 < Dd; k0 += 4) {
      v2f a  = *(const v2f*)(&qA[ln * 132 + k0 + koff]);
      v2f bb = __builtin_nontemporal_load((const v2f*)(kr + k0 + koff));
      if (!ktail)  // fused cache materialization (each element stored once)
        __builtin_nontemporal_store(bb, (v2f*)(kw + k0 + koff));
      c = wmma_f32(a, bb, c);
    }

    // ---- online softmax: only rows 0..3 (lg==0) carry real data ----
    const bool oob = (s0 + ln) >= SF;
    float pv[8];
    for (int r = 0; r < 8; ++r) pv[r] = 0.f;
    for (int r = 0; r < 4; ++r) {
      float sc   = oob ? -3.0e38f : c[r];
      float tmax = rowmax16(sc);              // valid within lg==0 group
      float mnew = fmaxf(mr[r], tmax);
      float resc = __expf(mr[r] - mnew);
      float p    = __expf(sc - mnew);
      float rs   = rowsum16(p);
      lr[r] = lr[r] * resc + rs;
      mr[r] = mnew;
      o[0][r] *= resc; o[1][r] *= resc; o[2][r] *= resc; o[3][r] *= resc;
      o[4][r] *= resc; o[5][r] *= resc; o[6][r] *= resc; o[7][r] *= resc;
      if (lg == 0) pv[r] = p;                 // P rows 4..15 stay 0
    }

    // ---- C-layout -> A-layout for P via per-wave LDS ----
    float* pw = &pS[wave][0];
    for (int r = 0; r < 8; ++r) pw[(r + 8 * lg) * 17 + ln] = pv[r];
    asm volatile("s_wait_dscnt 0" ::: "memory");   // intra-wave DS RAW

    // ---- O += P(16x16) . V(16x128), streaming V + fused store ----
    for (int k0 = 0; k0 < 16; k0 += 4) {
      v2f a;
      a.x = pw[ln * 17 + k0 + koff];
      a.y = pw[ln * 17 + k0 + koff + 1];
      int sA = s0 + k0 + koff;
      int sB = sA + 1;
      const bool tA = sA >= SC, tB = sB >= SC;
      if (sA > SF - 1) sA = SF - 1;
      if (sB > SF - 1) sB = SF - 1;
      const float* vrA = (tA ? (const float*)Vf : Vc) + (long)sA * Dd;
      const float* vrB = (tB ? (const float*)Vf : Vc) + (long)sB * Dd;
      float* vwA = Vf + (long)sA * Dd;
      float* vwB = Vf + (long)sB * Dd;
      for (int j = 0; j < 8; ++j) {
        float x = __builtin_nontemporal_load(vrA + j * 16 + ln);
        float y = __builtin_nontemporal_load(vrB + j * 16 + ln);
        if (!tA) __builtin_nontemporal_store(x, vwA + j * 16 + ln);
        if (!tB) __builtin_nontemporal_store(y, vwB + j * 16 + ln);
        v2f bb; bb.x = x; bb.y = y;
        o[j] = wmma_f32(a, bb, o[j]);
      }
    }
  }

  // ---- publish per-wave partials (rows 0..3 live in lg==0 lanes) ----
  if (lane == 0) {
    for (int r = 0; r < 4; ++r) { mS[wave][r] = mr[r]; lS[wave][r] = lr[r]; }
  }
  if (lg == 0) {
    for (int j = 0; j < 8; ++j)
      for (int r = 0; r < 4; ++r)
        accS[wave][r * 128 + j * 16 + ln] = o[j][r];
  }
  __syncthreads();

  // ---- combine 8 wave-partials; thread = d ----
  if (tid < 128) {
    const int d = tid;
    for (int row = 0; row < Tq; ++row) {
      float mt = -3.0e38f;
      for (int w = 0; w < NW; ++w) mt = fmaxf(mt, mS[w][row]);
      float lt = 0.f, ot = 0.f;
      for (int w = 0; w < NW; ++w) {
        float f = __expf(mS[w][row] - mt);
        lt += lS[w][row] * f;
        ot += accS[w][row * 128 + d] * f;
      }
      attn_out[((long)(b * Tq + row)) * HID + h * Dd + d] = ot / lt;
    }
  }
}

// ---------------------------------------------------------------------------
extern "C" void kernel_launch(void* const* d_in, const int* in_sizes, int n_in,
                              void* d_out, int out_size, void* d_ws, size_t ws_size,
                              hipStream_t stream) {
  const float* hidden  = (const float*)d_in[0];
  const float* cache_k = (const float*)d_in[1];
  const float* cache_v = (const float*)d_in[2];
  const float* cosb    = (const float*)d_in[3];
  const float* sinb    = (const float*)d_in[4];
  const float* Wq      = (const float*)d_in[5];
  const float* Wk      = (const float*)d_in[6];
  const float* Wv      = (const float*)d_in[7];
  const float* Wo      = (const float*)d_in[8];

  float* out   = (float*)d_out;                         // [8,4,4096]
  float* kfull = out + (long)NROW * HID;                // [8,32,4100,128]
  float* vfull = kfull + (long)Bq * Hh * SF * Dd;

  float* qbuf = (float*)d_ws;                           // [32][4096] roped q
  float* kbuf = qbuf + (long)NROW * HID;                // [32][4096] raw k
  float* abuf = kbuf + (long)NROW * HID;                // [32][4096] attn out

  // 1) projections (V goes straight into v_full tail)
  gemm_nt<<<32, 256, 0, stream>>>(hidden, Wq, qbuf, 0);
  gemm_nt<<<32, 256, 0, stream>>>(hidden, Wk, kbuf, 0);
  gemm_nt<<<32, 256, 0, stream>>>(hidden, Wv, vfull, 1);
  // 2) RoPE (q in place, k -> k_full tail)
  rope_append<<<256, 256, 0, stream>>>(qbuf, kbuf, cosb, sinb, kfull);
  // 3) fused flash attention + KV-cache materialization (single cache pass)
  attn_fused<<<Bq * Hh, 256, 0, stream>>>(qbuf, cache_k, cache_v,
                                          kfull, vfull, abuf);
  // 4) output projection
  gemm_nt<<<32, 256, 0, stream>>>(abuf, Wo, out, 0);
}